// TrainedDownsampling_1838246002902
// MI455X (gfx1250) — compile-verified
//
#include <hip/hip_runtime.h>
#include <stdint.h>
#include <math.h>

// Problem constants (from reference)
#define N_POINTS 131072   // 2^17
#define N_RETAIN 65536    // 2^16
#define BATCH    64
#define CHUNK    4096     // bitonic LDS chunk (2^12), 32KB of u64 keys per block

// ---------------------------------------------------------------------------
// CDNA5 async global->LDS copy (GLOBAL_LOAD_ASYNC_TO_LDS_B64, ASYNCcnt).
// Builtin signature (from clang diagnostic): first arg is v2i* in the global
// (__device__/AS1) address space, second is the LDS (AS3) destination.
// AS3 pointer value == low 32 bits of the flat address (ISA §10.2 apertures).
// ---------------------------------------------------------------------------
#if defined(__has_builtin)
#  if __has_builtin(__builtin_amdgcn_global_load_async_to_lds_b64)
#    define HAVE_ASYNC_LDS 1
#  endif
#endif
#ifndef HAVE_ASYNC_LDS
#  define HAVE_ASYNC_LDS 0
#endif

#if HAVE_ASYNC_LDS
typedef int v2i __attribute__((ext_vector_type(2)));
typedef __attribute__((address_space(1))) v2i gbl_v2i;
typedef __attribute__((address_space(3))) v2i lds_v2i;

__device__ static inline gbl_v2i* to_gbl_b64(const void* p) {
    return (gbl_v2i*)(uintptr_t)p;                 // flat global addr == AS1 addr
}
__device__ static inline lds_v2i* to_lds_b64(void* p) {
    return (lds_v2i*)(uint32_t)(uintptr_t)p;       // LDS offset = addr[31:0]
}
#endif

__device__ static inline void wait_async0() {
#if defined(__has_builtin) && __has_builtin(__builtin_amdgcn_s_wait_asynccnt)
    __builtin_amdgcn_s_wait_asynccnt(0);
#else
    asm volatile("s_wait_asynccnt 0x0" ::: "memory");
#endif
}

// Fill lk[0..CHUNK) from keys[base..base+CHUNK) using async-to-LDS when
// available (each of 1024 lanes issues 4 b64 transfers), else plain loads.
__device__ static inline void load_chunk_to_lds(uint64_t* lk, const uint64_t* __restrict__ keys,
                                                int base, int tid) {
#if HAVE_ASYNC_LDS
    for (int e = tid; e < CHUNK; e += 1024)
        __builtin_amdgcn_global_load_async_to_lds_b64(
            to_gbl_b64(keys + base + e), to_lds_b64(&lk[e]), 0, 0);
    wait_async0();
#else
    for (int e = tid; e < CHUNK; e += 1024) lk[e] = keys[base + e];
#endif
    __syncthreads();
}

// ---------------------------------------------------------------------------
// Softmax reductions (fixed-shape, deterministic)
// ---------------------------------------------------------------------------
__global__ void k_partial_max(const float* __restrict__ w, float* __restrict__ partials) {
    __shared__ float sm[256];
    int tid = threadIdx.x;
    int gid = blockIdx.x * 256 + tid;
    float m = -INFINITY;
    for (int i = gid; i < N_POINTS; i += 256 * 256) m = fmaxf(m, w[i]);
    sm[tid] = m;
    __syncthreads();
    for (int s = 128; s > 0; s >>= 1) {
        if (tid < s) sm[tid] = fmaxf(sm[tid], sm[tid + s]);
        __syncthreads();
    }
    if (tid == 0) partials[blockIdx.x] = sm[0];
}

__global__ void k_final_max(const float* __restrict__ partials, float* __restrict__ out) {
    __shared__ float sm[256];
    int tid = threadIdx.x;
    sm[tid] = partials[tid];
    __syncthreads();
    for (int s = 128; s > 0; s >>= 1) {
        if (tid < s) sm[tid] = fmaxf(sm[tid], sm[tid + s]);
        __syncthreads();
    }
    if (tid == 0) out[0] = sm[0];
}

__global__ void k_exp_partial_sum(const float* __restrict__ w, const float* __restrict__ scal,
                                  float* __restrict__ expv, float* __restrict__ partials) {
    __shared__ float sm[256];
    float M = scal[0];
    int tid = threadIdx.x;
    int gid = blockIdx.x * 256 + tid;
    float s = 0.f;
    for (int i = gid; i < N_POINTS; i += 256 * 256) {
        float e = expf(w[i] - M);
        expv[i] = e;
        s += e;
    }
    sm[tid] = s;
    __syncthreads();
    for (int st = 128; st > 0; st >>= 1) {
        if (tid < st) sm[tid] += sm[tid + st];
        __syncthreads();
    }
    if (tid == 0) partials[blockIdx.x] = sm[0];
}

__global__ void k_final_sum(const float* __restrict__ partials, float* __restrict__ out) {
    __shared__ float sm[256];
    int tid = threadIdx.x;
    sm[tid] = partials[tid];
    __syncthreads();
    for (int st = 128; st > 0; st >>= 1) {
        if (tid < st) sm[tid] += sm[tid + st];
        __syncthreads();
    }
    if (tid == 0) out[0] = sm[0];
}

// sum of probs = sum(expv[i] / S)  (reference sums the divided values)
__global__ void k_prob_partial_sum(const float* __restrict__ expv, const float* __restrict__ scal,
                                   float* __restrict__ partials) {
    __shared__ float sm[256];
    float S = scal[1];
    int tid = threadIdx.x;
    int gid = blockIdx.x * 256 + tid;
    float s = 0.f;
    for (int i = gid; i < N_POINTS; i += 256 * 256) s += expv[i] / S;
    sm[tid] = s;
    __syncthreads();
    for (int st = 128; st > 0; st >>= 1) {
        if (tid < st) sm[tid] += sm[tid + st];
        __syncthreads();
    }
    if (tid == 0) partials[blockIdx.x] = sm[0];
}

// ---------------------------------------------------------------------------
// Key build: probs > 0, so float bit pattern is order-monotonic.
// key = bits(prob) << 32 | ~index  -> descending u64 sort gives descending
// prob with ascending-index tie-break (jax.lax.top_k semantics).
// ---------------------------------------------------------------------------
__global__ void k_build_keys(const float* __restrict__ expv, const float* __restrict__ scal,
                             uint64_t* __restrict__ keys) {
    int i = blockIdx.x * 256 + threadIdx.x;   // grid covers N exactly
    float p = expv[i] / scal[1];
    uint32_t pb = __float_as_uint(p);
    keys[i] = ((uint64_t)pb << 32) | (uint32_t)(~(uint32_t)i);
}

// ---------------------------------------------------------------------------
// Bitonic sort, descending overall. desc block iff ((global_index & k) == 0).
// ---------------------------------------------------------------------------
__global__ __launch_bounds__(1024) void k_bitonic_local_sort(uint64_t* __restrict__ keys) {
    __shared__ uint64_t lk[CHUNK];
    int tid = threadIdx.x;
    int base = blockIdx.x * CHUNK;
    load_chunk_to_lds(lk, keys, base, tid);
    for (int k = 2; k <= CHUNK; k <<= 1) {
        for (int j = k >> 1; j >= 1; j >>= 1) {
            for (int p = tid; p < CHUNK / 2; p += 1024) {
                int l = ((p & ~(j - 1)) << 1) | (p & (j - 1));
                int r = l + j;
                bool desc = (((base + l) & k) == 0);
                uint64_t a = lk[l], b = lk[r];
                if (desc ? (a < b) : (a > b)) { lk[l] = b; lk[r] = a; }
            }
            __syncthreads();
        }
    }
    for (int e = tid; e < CHUNK; e += 1024) keys[base + e] = lk[e];
}

__global__ void k_bitonic_global_pass(uint64_t* __restrict__ keys, int k, int j) {
    int p = blockIdx.x * 256 + threadIdx.x;   // N/2 pairs total
    int l = ((p & ~(j - 1)) << 1) | (p & (j - 1));
    int r = l + j;
    bool desc = ((l & k) == 0);
    uint64_t a = keys[l], b = keys[r];
    if (desc ? (a < b) : (a > b)) { keys[l] = b; keys[r] = a; }
}

__global__ __launch_bounds__(1024) void k_bitonic_local_merge(uint64_t* __restrict__ keys, int k) {
    __shared__ uint64_t lk[CHUNK];
    int tid = threadIdx.x;
    int base = blockIdx.x * CHUNK;
    load_chunk_to_lds(lk, keys, base, tid);
    for (int j = CHUNK >> 1; j >= 1; j >>= 1) {
        for (int p = tid; p < CHUNK / 2; p += 1024) {
            int l = ((p & ~(j - 1)) << 1) | (p & (j - 1));
            int r = l + j;
            bool desc = (((base + l) & k) == 0);
            uint64_t a = lk[l], b = lk[r];
            if (desc ? (a < b) : (a > b)) { lk[l] = b; lk[r] = a; }
        }
        __syncthreads();
    }
    for (int e = tid; e < CHUNK; e += 1024) keys[base + e] = lk[e];
}

// ---------------------------------------------------------------------------
// Gather: out[b][r] = points[b][idx(r)] * prob(r) / (S2 + 1e-6)
// One float4 row per thread; coalesced b128 writes; key stream prefetched
// (emits gfx1250 global_prefetch_b8).
// ---------------------------------------------------------------------------
__global__ void k_gather(const float4* __restrict__ points, const uint64_t* __restrict__ keys,
                         const float* __restrict__ scal, float4* __restrict__ out) {
    int t = blockIdx.x * 256 + threadIdx.x;   // [0, BATCH*N_RETAIN)
    int b = t >> 16;                          // / N_RETAIN
    int r = t & (N_RETAIN - 1);
    // keys array is N_POINTS long; r + 2048 < N_POINTS, always in-bounds.
    __builtin_prefetch(&keys[r + 2048], 0, 1);
    uint64_t key = keys[r];
    uint32_t idx = ~(uint32_t)key;                       // recover original index
    float prob = __uint_as_float((uint32_t)(key >> 32)); // sampling prob of idx
    float denom = scal[2] + 1e-6f;                       // sum(probs) + 1e-6
    float s = prob / denom;
    float4 pt = points[b * N_POINTS + (int)idx];
    out[t] = make_float4(pt.x * s, pt.y * s, pt.z * s, pt.w * s);
}

// ---------------------------------------------------------------------------
extern "C" void kernel_launch(void* const* d_in, const int* in_sizes, int n_in,
                              void* d_out, int out_size, void* d_ws, size_t ws_size,
                              hipStream_t stream) {
    (void)in_sizes; (void)n_in; (void)out_size; (void)ws_size;
    const float* points = (const float*)d_in[0];   // [64, 131072, 4] f32
    const float* w      = (const float*)d_in[1];   // [131072] f32
    float* out = (float*)d_out;                    // [64, 65536, 4] f32

    // Workspace layout (~1.5 MB):
    //   expv     : N floats                 @ 0
    //   keys     : N u64                    @ N*4       (8-aligned)
    //   partials : 256 floats               @ N*4 + N*8
    //   scal     : [0]=max, [1]=S, [2]=S2   after partials
    char* wsb = (char*)d_ws;
    float*    expv     = (float*)wsb;
    uint64_t* keys     = (uint64_t*)(wsb + (size_t)N_POINTS * 4);
    float*    partials = (float*)(wsb + (size_t)N_POINTS * 4 + (size_t)N_POINTS * 8);
    float*    scal     = partials + 256;

    // 1) softmax statistics
    k_partial_max<<<256, 256, 0, stream>>>(w, partials);
    k_final_max<<<1, 256, 0, stream>>>(partials, scal + 0);
    k_exp_partial_sum<<<256, 256, 0, stream>>>(w, scal, expv, partials);
    k_final_sum<<<1, 256, 0, stream>>>(partials, scal + 1);
    k_prob_partial_sum<<<256, 256, 0, stream>>>(expv, scal, partials);
    k_final_sum<<<1, 256, 0, stream>>>(partials, scal + 2);

    // 2) build sort keys (prob bits | ~index)
    k_build_keys<<<N_POINTS / 256, 256, 0, stream>>>(expv, scal, keys);

    // 3) bitonic sort, descending: LDS for k<=4096, global passes for j>=4096,
    //    LDS tail merge for j<=2048.
    k_bitonic_local_sort<<<N_POINTS / CHUNK, 1024, 0, stream>>>(keys);
    for (int k = CHUNK * 2; k <= N_POINTS; k <<= 1) {
        for (int j = k >> 1; j >= CHUNK; j >>= 1)
            k_bitonic_global_pass<<<N_POINTS / 512, 256, 0, stream>>>(keys, k, j);
        k_bitonic_local_merge<<<N_POINTS / CHUNK, 1024, 0, stream>>>(keys, k);
    }

    // 4) batched gather + scale (bandwidth-bound: ~128 MB -> ~5.5 us floor)
    k_gather<<<(BATCH * N_RETAIN) / 256, 256, 0, stream>>>(
        (const float4*)points, keys, scal, (float4*)out);
}